// MessagePassing_56908316672261
// MI455X (gfx1250) — compile-verified
//
#include <hip/hip_runtime.h>
#include <math.h>

typedef float v2f __attribute__((ext_vector_type(2)));
typedef float v8f __attribute__((ext_vector_type(8)));

#define WPB 4            // waves per block
#define THREADS (WPB*32)
#define MT 2             // M-tiles (of 16 edges) per wave iteration
#define W1_STRIDE 288    // padded k-pair row stride (floats) -> conflict-free b64

__device__ __forceinline__ v8f wmma_f32_16x16x4(v2f a, v2f b, v8f c) {
  // D = A(16x4) * B(4x16) + C(16x16), fp32 matrix pipe on gfx1250
  return __builtin_amdgcn_wmma_f32_16x16x4_f32(false, a, false, b, (short)0, c,
                                               false, false);
}

// ---------------------------------------------------------------------------
// Per-edge pipeline: weight MLP (2 WMMA GEMMs) -> tp_uvu -> scatter + edge out
// One wave owns 32 edges (2 tiles of 16) per iteration.
// ---------------------------------------------------------------------------
__global__ __launch_bounds__(THREADS)
void edge_kernel(const float* __restrict__ x,      // [N,32] lin1 output
                 const float* __restrict__ ea,     // [E,4]  current edge_attr
                 const float* __restrict__ es,     // [E,16] edge_scalars
                 const int*   __restrict__ src,
                 const int*   __restrict__ dst,
                 const float* __restrict__ w0,     // [16,64]
                 const float* __restrict__ w1,     // [64,128]
                 const float* __restrict__ scw,    // [32,4,4]
                 float* __restrict__ agg,          // [N,32] zero-initialized
                 float* __restrict__ ea_out,       // [E,4]
                 int n_tiles, int E)
{
  // fc_w1 staged k-pair interleaved: element (kr, col) at
  //   lds_w1[(kr>>1)*W1_STRIDE + col*2 + (kr&1)]
  // so a B-fragment (kr, kr+1 @ col) is one aligned ds_load_b64.
  __shared__ float lds_w1[32*W1_STRIDE];           // 36 KB
  __shared__ float lds_sce[512];                   // 2 KB
  __shared__ float lds_wv[WPB][16*68];             // per-wave h / ef staging

  const int tid  = threadIdx.x;
  const int lane = tid & 31;
  const int wave = tid >> 5;
  const int half = lane >> 4;      // 0: lanes 0-15, 1: lanes 16-31
  const int l16  = lane & 15;

  for (int i = tid; i < 64*128; i += THREADS) {
    const int kr = i >> 7, col = i & 127;
    lds_w1[(kr >> 1)*W1_STRIDE + col*2 + (kr & 1)] = w1[i] * 0.125f; // 1/sqrt(64)
  }
  for (int i = tid; i < 512; i += THREADS) lds_sce[i] = scw[i];
  __syncthreads();

  // fc_w0 B-fragments (K=4 x N=16 tiles), scaled by 1/sqrt(16)
  v2f b0[4][4];
#pragma unroll
  for (int n = 0; n < 4; ++n)
#pragma unroll
    for (int kk = 0; kk < 4; ++kk) {
      const int kr = kk*4 + half*2;
      b0[n][kk][0] = w0[(kr+0)*64 + n*16 + l16] * 0.25f;
      b0[n][kk][1] = w0[(kr+1)*64 + n*16 + l16] * 0.25f;
    }

  float* hbuf = &lds_wv[wave][0];

  for (int t = blockIdx.x*WPB + wave; t < n_tiles; t += gridDim.x*WPB) {
    const int e0 = t*32;

    // ---------------- GEMM1 per M-tile: h = sin( es @ w0 / 4 ) -------------
    // Re-layout C->A through the per-wave LDS tile, reused for both M-tiles.
    v2f a2[MT][16];
#pragma unroll
    for (int m = 0; m < MT; ++m) {
      const int e0m = e0 + m*16;
      int erow = e0m + l16; if (erow >= E) erow = E - 1;
      v2f a1[4];
#pragma unroll
      for (int kk = 0; kk < 4; ++kk) {
        const float* p = es + (size_t)erow*16 + kk*4 + half*2;
        a1[kk][0] = p[0]; a1[kk][1] = p[1];
      }
      v8f c1[4];
#pragma unroll
      for (int n = 0; n < 4; ++n) {
        v8f c = {};
#pragma unroll
        for (int kk = 0; kk < 4; ++kk)
          c = wmma_f32_16x16x4(a1[kk], b0[n][kk], c);
        c1[n] = c;
      }
      // sin + spill h tile (row M = j + 8*half, col = 16n + l16)
#pragma unroll
      for (int n = 0; n < 4; ++n)
#pragma unroll
        for (int j = 0; j < 8; ++j)
          hbuf[(j + 8*half)*68 + n*16 + l16] = __sinf(c1[n][j]);
      asm volatile("s_wait_dscnt 0x0" ::: "memory");
      // gather A-fragments for GEMM2
#pragma unroll
      for (int k = 0; k < 16; ++k) {
        const float* p = hbuf + l16*68 + k*4 + half*2;
        a2[m][k][0] = p[0]; a2[m][k][1] = p[1];
      }
      asm volatile("s_wait_dscnt 0x0" ::: "memory");
    }

    // ---------------- GEMM2: w = h @ (w1 / 8), B reused across M-tiles -----
    v8f c2[MT][8];
#pragma unroll
    for (int n = 0; n < 8; ++n) {
      v8f cA = {}, cB = {};
#pragma unroll
      for (int k = 0; k < 16; ++k) {
        const v2f bb = *(const v2f*)(lds_w1 + (2*k + half)*W1_STRIDE
                                             + (n*16 + l16)*2);
        cA = wmma_f32_16x16x4(a2[0][k], bb, cA);
        cB = wmma_f32_16x16x4(a2[1][k], bb, cB);
      }
      c2[0][n] = cA; c2[1][n] = cB;
    }

    // -------- per M-tile: tp_uvu -> scatter-add -> edge_attr output --------
    float* efbuf = hbuf;              // reuse staging (stride 36)
    const int g4 = l16 >> 2;
#pragma unroll
    for (int m = 0; m < MT; ++m) {
      const int e0m = e0 + m*16;
      // c2 element: row = j + 8*half, col = 16n + l16; u = col/4, v = col&3
#pragma unroll
      for (int j = 0; j < 8; ++j) {
        const int row = j + 8*half;
        const int e = e0m + row;
        const bool valid = e < E;
        const int ec = valid ? e : E - 1;
        const float eav = ea[(size_t)ec*4 + (lane & 3)];
        const int sj = src[ec];
        const int dj = dst[ec];
#pragma unroll
        for (int n = 0; n < 8; ++n) {
          const int u = n*4 + g4;
          float p = c2[m][n][j] * eav;
          p += __shfl_xor(p, 1, 32);          // butterfly sum over v (4 lanes)
          p += __shfl_xor(p, 2, 32);
          const float ef = p * x[(size_t)sj*32 + u] * 0.5f;
          if (((lane & 3) == 0) && valid) {
            atomicAdd(&agg[(size_t)dj*32 + u], ef);  // segment_sum scatter
            efbuf[row*36 + u] = ef;
          }
        }
      }
      asm volatile("s_wait_dscnt 0x0" ::: "memory");

      // edge_out = ea + fctp(ef, ea, sce_w)/(sqrt(128)*4)
      {
        const int e  = lane >> 1;        // 16 edges x 2 lanes
        const int kk = (lane & 1) * 2;   // each lane does 2 of 4 outputs
        const int eg = e0m + e;
        const bool v = eg < E;
        const int ec = v ? eg : E - 1;
        const float e0v = ea[(size_t)ec*4+0], e1v = ea[(size_t)ec*4+1];
        const float e2v = ea[(size_t)ec*4+2], e3v = ea[(size_t)ec*4+3];
        float acc0 = 0.f, acc1 = 0.f;
#pragma unroll 4
        for (int u = 0; u < 32; ++u) {
          const float efv = efbuf[e*36 + u];
          const float* sw = &lds_sce[u*16];          // sce_w[u][v][k]
          acc0 += efv * (e0v*sw[kk]   + e1v*sw[4+kk] + e2v*sw[8+kk] + e3v*sw[12+kk]);
          acc1 += efv * (e0v*sw[kk+1] + e1v*sw[5+kk] + e2v*sw[9+kk] + e3v*sw[13+kk]);
        }
        const float s = 0.02209708691f;  // 1/(sqrt(32*4)*sqrt(16))
        if (v) {
          ea_out[(size_t)eg*4 + kk]     = (kk == 0 ? e0v : e2v) + acc0*s;
          ea_out[(size_t)eg*4 + kk + 1] = (kk == 0 ? e1v : e3v) + acc1*s;
        }
      }
      asm volatile("s_wait_dscnt 0x0" ::: "memory");
    }
  }
}

// ---------------------------------------------------------------------------
// Per-node pre: node_sc = fctp(nf, attr, sc_w), x = fctp(nf, attr, lin1_w)
// One wave per node, lane = output channel k.
// ---------------------------------------------------------------------------
__global__ __launch_bounds__(128)
void node_pre_kernel(const float* __restrict__ nf, const float* __restrict__ attr,
                     const float* __restrict__ scw, const float* __restrict__ l1w,
                     float* __restrict__ x, float* __restrict__ nsc, int N)
{
  const int w = (blockIdx.x*blockDim.x + threadIdx.x) >> 5;
  const int lane = threadIdx.x & 31;
  if (w >= N) return;
  float av[8];
#pragma unroll
  for (int v = 0; v < 8; ++v) av[v] = attr[(size_t)w*8 + v];
  float asc = 0.f, ax = 0.f;
  for (int u = 0; u < 32; ++u) {
    const float nfv = nf[(size_t)w*32 + u];
#pragma unroll
    for (int v = 0; v < 8; ++v) {
      const float tt = nfv * av[v];
      const int idx = (u*8 + v)*32 + lane;
      asc += tt * scw[idx];
      ax  += tt * l1w[idx];
    }
  }
  nsc[(size_t)w*32 + lane] = asc * 0.0625f;   // 1/sqrt(32*8)
  x  [(size_t)w*32 + lane] = ax  * 0.0625f;
}

// ---------------------------------------------------------------------------
// Per-node post: nf_out = node_sc + alpha * fctp(agg/4, attr, lin2_w)
// ---------------------------------------------------------------------------
__global__ __launch_bounds__(128)
void node_post_kernel(const float* __restrict__ agg, const float* __restrict__ attr,
                      const float* __restrict__ l2w, const float* __restrict__ aw,
                      const float* __restrict__ nsc, float* __restrict__ out,
                      int N, int apply_sin)
{
  const int w = (blockIdx.x*blockDim.x + threadIdx.x) >> 5;
  const int lane = threadIdx.x & 31;
  if (w >= N) return;
  float av[8];
#pragma unroll
  for (int v = 0; v < 8; ++v) av[v] = attr[(size_t)w*8 + v];
  float ac = 0.f, aa = 0.f;
  for (int u = 0; u < 32; ++u) {
    const float gv = agg[(size_t)w*32 + u] * 0.25f;   // 1/sqrt(NUM_NEIGHBORS)
#pragma unroll
    for (int v = 0; v < 8; ++v) {
      const float tt = gv * av[v];
      ac += tt * l2w[(u*8 + v)*32 + lane];
      aa += tt * aw[u*8 + v];
    }
  }
  float o = nsc[(size_t)w*32 + lane] + (aa * 0.0625f) * (ac * 0.0625f);
  if (apply_sin) o = __sinf(o);
  out[(size_t)w*32 + lane] = o;
}

// ---------------------------------------------------------------------------
extern "C" void kernel_launch(void* const* d_in, const int* in_sizes, int n_in,
                              void* d_out, int out_size, void* d_ws, size_t ws_size,
                              hipStream_t stream) {
  const float* nf0     = (const float*)d_in[0];
  const float* attr    = (const float*)d_in[1];
  const int*   esrc    = (const int*)  d_in[2];
  const int*   edst    = (const int*)  d_in[3];
  const float* ea0     = (const float*)d_in[4];
  const float* escal   = (const float*)d_in[5];
  const float* sc_w    = (const float*)d_in[6];
  const float* lin1_w  = (const float*)d_in[7];
  const float* lin2_w  = (const float*)d_in[8];
  const float* alpha_w = (const float*)d_in[9];
  const float* sce_w   = (const float*)d_in[10];
  const float* fc_w0   = (const float*)d_in[11];
  const float* fc_w1   = (const float*)d_in[12];

  const int N = in_sizes[0] / 32;
  const int E = in_sizes[4] / 4;

  float* ws  = (float*)d_ws;
  float* x   = ws;
  float* nsc = ws + (size_t)N*32;
  float* agg = ws + 2*(size_t)N*32;
  float* nf1 = ws + 3*(size_t)N*32;
  float* ea1 = ws + 4*(size_t)N*32;

  float* outnf = (float*)d_out;
  float* outea = (float*)d_out + (size_t)N*32;

  const int n_tiles     = (E + 31) / 32;         // 32 edges per wave iteration
  const int node_blocks = (N + 3) / 4;
  int edge_blocks = (n_tiles + WPB - 1) / WPB;
  if (edge_blocks > 4096) edge_blocks = 4096;
  if (edge_blocks < 1) edge_blocks = 1;

  for (int l = 0; l < 2; ++l) {
    const float* nf_in  = (l == 0) ? nf0 : nf1;
    const float* ea_in  = (l == 0) ? ea0 : ea1;
    float*       nf_out = (l == 0) ? nf1 : outnf;
    float*       ea_out = (l == 0) ? ea1 : outea;

    hipMemsetAsync(agg, 0, (size_t)N*32*sizeof(float), stream);

    node_pre_kernel<<<node_blocks, 128, 0, stream>>>(
        nf_in, attr, sc_w + (size_t)l*8192, lin1_w + (size_t)l*8192, x, nsc, N);

    edge_kernel<<<edge_blocks, THREADS, 0, stream>>>(
        x, ea_in, escal, esrc, edst,
        fc_w0 + (size_t)l*1024, fc_w1 + (size_t)l*8192, sce_w + (size_t)l*512,
        agg, ea_out, n_tiles, E);

    node_post_kernel<<<node_blocks, 128, 0, stream>>>(
        agg, attr, lin2_w + (size_t)l*8192, alpha_w + (size_t)l*256,
        nsc, nf_out, N, (l == 0) ? 1 : 0);
  }
}